// GAT_4758823764090
// MI455X (gfx1250) — compile-verified
//
#include <hip/hip_runtime.h>
#include <hip/hip_bf16.h>

// ---------------------------------------------------------------------------
// 2-layer GAT for MI455X (gfx1250, wave32).
// GEMMs use V_WMMA_F32_16X16X4_F32 (f32-exact matrix pipe).
// Edge phase uses int-encoded float atomic-max + f32 atomic adds (L2 atomics).
// ---------------------------------------------------------------------------

typedef float v2f __attribute__((ext_vector_type(2)));
typedef float v8f __attribute__((ext_vector_type(8)));

#define IN_DIM   256
#define HEADS1   6
#define HID1     64
#define F1       (HEADS1 * HID1)   // 384
#define OUT_DIM  32
#define NEG_SLOPE 0.2f

// monotonic order-preserving float<->int encoding (for integer atomicMax)
__device__ __forceinline__ int enc_f32(float f) {
    int i = __float_as_int(f);
    return i < 0 ? (i ^ 0x7FFFFFFF) : i;
}
__device__ __forceinline__ float dec_f32(int i) {
    return __int_as_float(i < 0 ? (i ^ 0x7FFFFFFF) : i);
}
#define ENC_NEG_INF ((int)0x807FFFFF)   // enc_f32(-inf)

__global__ void fill_f32_kernel(float* __restrict__ p, float v, int n) {
    int i = blockIdx.x * blockDim.x + threadIdx.x;
    if (i < n) p[i] = v;
}
__global__ void fill_i32_kernel(int* __restrict__ p, int v, int n) {
    int i = blockIdx.x * blockDim.x + threadIdx.x;
    if (i < n) p[i] = v;
}

// ---------------------------------------------------------------------------
// C[M,Nout] = A[M,K] @ B[K,Nout], all f32 row-major.
// One wave computes one 16x16 tile via V_WMMA_F32_16X16X4_F32.
// Block = NT waves (NT = blockDim.x/32) covering NT adjacent N-tiles of one
// M-tile; A tile (16 x K) staged in LDS (row stride 388 => conflict-free).
// ---------------------------------------------------------------------------
__global__ void gemm_wmma_f32(const float* __restrict__ A,
                              const float* __restrict__ B,
                              float* __restrict__ C,
                              int M, int K, int Nout) {
    __shared__ float As[16 * 388];           // supports K <= 384
    const int tid   = threadIdx.x;
    const int lane  = tid & 31;
    const int wave  = tid >> 5;
    const int NT    = blockDim.x >> 5;
    const int mBase = blockIdx.x * 16;
    const int nBase = (blockIdx.y * NT + wave) * 16;

    // cooperative load of the 16 x K A-tile into LDS (coalesced along K)
    for (int idx = tid; idx < 16 * K; idx += blockDim.x) {
        int r = idx / K;
        int c = idx - r * K;
        int gr = mBase + r;
        As[r * 388 + c] = (gr < M) ? A[(long)gr * K + c] : 0.0f;
    }
    __syncthreads();

    v8f acc = {0.f, 0.f, 0.f, 0.f, 0.f, 0.f, 0.f, 0.f};

    // A frag: lanes 0-15 hold row=lane, K = kk,kk+1 ; lanes 16-31 K = kk+2,kk+3
    const int ar   = lane & 15;
    const int koff = (lane >> 4) << 1;
    const int col  = nBase + (lane & 15);
    const float* Bcol = B + col;

    #pragma unroll 4
    for (int kk = 0; kk < K; kk += 4) {
        v2f a, b;
        a.x = As[ar * 388 + kk + koff];
        a.y = As[ar * 388 + kk + koff + 1];
        b.x = Bcol[(long)(kk + koff) * Nout];
        b.y = Bcol[(long)(kk + koff + 1) * Nout];
        acc = __builtin_amdgcn_wmma_f32_16x16x4_f32(
            false, a, false, b, (short)0, acc, false, false);
    }

    // D layout: VGPR v -> M = v + 8*(lane>=16), N = lane&15
    const int mOff = (lane >> 4) << 3;
    #pragma unroll
    for (int v = 0; v < 8; ++v) {
        int gr = mBase + mOff + v;
        if (gr < M && col < Nout) C[(long)gr * Nout + col] = acc[v];
    }
}

// ---------------------------------------------------------------------------
// a_src[n,h] = sum_c H[n,h,c]*att_src[h,c]  (same for a_dst).
// One wave per (node, head); wave32 shuffle reduction.
// ---------------------------------------------------------------------------
__global__ void attn_coef_kernel(const float* __restrict__ H,
                                 const float* __restrict__ att_s,
                                 const float* __restrict__ att_d,
                                 float* __restrict__ a_s,
                                 float* __restrict__ a_d,
                                 int nNodes, int heads, int ch) {
    int wid  = (blockIdx.x * blockDim.x + threadIdx.x) >> 5;
    int lane = threadIdx.x & 31;
    if (wid >= nNodes * heads) return;
    int n = wid / heads;
    int h = wid - n * heads;
    const float* row = H + (long)n * heads * ch + h * ch;
    float ss = 0.f, sd = 0.f;
    for (int c = lane; c < ch; c += 32) {
        float v = row[c];
        ss += v * att_s[h * ch + c];
        sd += v * att_d[h * ch + c];
    }
    #pragma unroll
    for (int off = 16; off > 0; off >>= 1) {
        ss += __shfl_xor(ss, off);
        sd += __shfl_xor(sd, off);
    }
    if (lane == 0) { a_s[wid] = ss; a_d[wid] = sd; }
}

__device__ __forceinline__ void edge_sd(const int* __restrict__ ei, int E,
                                        int t, int& s, int& d) {
    if (t < E) { s = ei[t]; d = ei[E + t]; }
    else       { s = t - E; d = t - E; }   // self loops appended
}

// per-edge LeakyReLU attention logit -> segment max (int-encoded atomicMax)
__global__ void edge_max_kernel(const int* __restrict__ ei, int E, int eTot,
                                const float* __restrict__ a_s,
                                const float* __restrict__ a_d,
                                int heads, int* __restrict__ mEnc) {
    int t = blockIdx.x * blockDim.x + threadIdx.x;
    if (t >= eTot) return;
    int s, d; edge_sd(ei, E, t, s, d);
    for (int h = 0; h < heads; ++h) {
        float v = a_s[s * heads + h] + a_d[d * heads + h];
        v = v > 0.f ? v : NEG_SLOPE * v;
        atomicMax(&mEnc[d * heads + h], enc_f32(v));
    }
}

// ex = exp(e - m[dst]); denom[dst] += ex ; ex stored per edge
__global__ void edge_expsum_kernel(const int* __restrict__ ei, int E, int eTot,
                                   const float* __restrict__ a_s,
                                   const float* __restrict__ a_d,
                                   int heads,
                                   const int* __restrict__ mEnc,
                                   float* __restrict__ exBuf,
                                   float* __restrict__ denom) {
    int t = blockIdx.x * blockDim.x + threadIdx.x;
    if (t >= eTot) return;
    int s, d; edge_sd(ei, E, t, s, d);
    for (int h = 0; h < heads; ++h) {
        float v = a_s[s * heads + h] + a_d[d * heads + h];
        v = v > 0.f ? v : NEG_SLOPE * v;
        float ex = __expf(v - dec_f32(mEnc[d * heads + h]));
        exBuf[(long)t * heads + h] = ex;
        atomicAdd(&denom[d * heads + h], ex);
    }
}

// out[dst, c] += H[src, c] * (ex[e, c/ch] / denom[dst, c/ch])
// blockDim.x = epb * (heads*ch); epb edges per block.
__global__ void edge_aggregate_kernel(const int* __restrict__ ei, int E, int eTot,
                                      const float* __restrict__ H,
                                      const float* __restrict__ exBuf,
                                      const float* __restrict__ denom,
                                      float* __restrict__ out,
                                      int heads, int ch) {
    int F   = heads * ch;
    int epb = blockDim.x / F;
    int le  = threadIdx.x / F;
    int c   = threadIdx.x - le * F;
    int t   = blockIdx.x * epb + le;
    if (t >= eTot) return;
    int s, d; edge_sd(ei, E, t, s, d);
    int h = c / ch;
    float alpha = exBuf[(long)t * heads + h] / denom[d * heads + h];
    atomicAdd(&out[(long)d * F + c], H[(long)s * F + c] * alpha);
}

// X[i] = (X[i] + b[i % F]), optional ReLU, in place
__global__ void bias_act_kernel(float* __restrict__ X,
                                const float* __restrict__ b,
                                int n, int F, int doRelu) {
    int i = blockIdx.x * blockDim.x + threadIdx.x;
    if (i >= n) return;
    float v = X[i] + b[i % F];
    if (doRelu) v = v > 0.f ? v : 0.f;
    X[i] = v;
}

// ---------------------------------------------------------------------------
extern "C" void kernel_launch(void* const* d_in, const int* in_sizes, int n_in,
                              void* d_out, int out_size, void* d_ws, size_t ws_size,
                              hipStream_t stream) {
    const float* x        = (const float*)d_in[0];
    const int*   ei       = (const int*)  d_in[1];   // [2, E]
    const float* W1       = (const float*)d_in[2];   // [256, 384]
    const float* att_src1 = (const float*)d_in[3];   // [6*64]
    const float* att_dst1 = (const float*)d_in[4];
    const float* b1       = (const float*)d_in[5];   // [384]
    const float* W2       = (const float*)d_in[6];   // [384, 32]
    const float* att_src2 = (const float*)d_in[7];   // [32]
    const float* att_dst2 = (const float*)d_in[8];
    const float* b2       = (const float*)d_in[9];   // [32]

    const int N    = in_sizes[0] / IN_DIM;   // 50000
    const int E    = in_sizes[1] / 2;        // 800000
    const int eTot = E + N;                  // with self loops

    // ---- workspace carve-out ----
    char*  ws  = (char*)d_ws;
    size_t off = 0;
    auto carve = [&](size_t bytes) -> void* {
        void* p = ws + off;
        off += (bytes + 255) & ~(size_t)255;
        return p;
    };
    float* H1    = (float*)carve((size_t)N * F1 * 4);
    float* OUT1  = (float*)carve((size_t)N * F1 * 4);   // also h1b after bias+relu
    float* AS1   = (float*)carve((size_t)N * HEADS1 * 4);
    float* AD1   = (float*)carve((size_t)N * HEADS1 * 4);
    int*   M1    = (int*)  carve((size_t)N * HEADS1 * 4);
    float* DEN1  = (float*)carve((size_t)N * HEADS1 * 4);
    float* EX1   = (float*)carve((size_t)eTot * HEADS1 * 4);
    float* H2    = (float*)carve((size_t)N * OUT_DIM * 4);
    float* AS2   = (float*)carve((size_t)N * 4);
    float* AD2   = (float*)carve((size_t)N * 4);
    int*   M2    = (int*)  carve((size_t)N * 4);
    float* DEN2  = (float*)carve((size_t)N * 4);
    float* EX2   = (float*)carve((size_t)eTot * 4);
    (void)ws_size;

    float* OUT2 = (float*)d_out;             // [N, 32] accumulated in place

    const int mTiles = (N + 15) / 16;        // 3125
    const int TB = 256;
    auto blocks = [&](int n) { return (n + TB - 1) / TB; };

    // ================= Layer 1 =================
    // H1 = x @ W1   (M=N, K=256, Nout=384) : 8 waves/block -> 3 N-groups
    gemm_wmma_f32<<<dim3(mTiles, F1 / (16 * 8)), 256, 0, stream>>>(
        x, W1, H1, N, IN_DIM, F1);

    attn_coef_kernel<<<blocks(N * HEADS1 * 32), TB, 0, stream>>>(
        H1, att_src1, att_dst1, AS1, AD1, N, HEADS1, HID1);

    fill_i32_kernel<<<blocks(N * HEADS1), TB, 0, stream>>>(M1, ENC_NEG_INF, N * HEADS1);
    fill_f32_kernel<<<blocks(N * HEADS1), TB, 0, stream>>>(DEN1, 0.f, N * HEADS1);
    fill_f32_kernel<<<blocks(N * F1), TB, 0, stream>>>(OUT1, 0.f, N * F1);

    edge_max_kernel<<<blocks(eTot), TB, 0, stream>>>(ei, E, eTot, AS1, AD1, HEADS1, M1);
    edge_expsum_kernel<<<blocks(eTot), TB, 0, stream>>>(
        ei, E, eTot, AS1, AD1, HEADS1, M1, EX1, DEN1);

    // one edge per block of 384 threads (6 heads x 64 ch)
    edge_aggregate_kernel<<<eTot, F1, 0, stream>>>(
        ei, E, eTot, H1, EX1, DEN1, OUT1, HEADS1, HID1);

    bias_act_kernel<<<blocks(N * F1), TB, 0, stream>>>(OUT1, b1, N * F1, F1, 1);

    // ================= Layer 2 =================
    // H2 = relu(h1) @ W2  (M=N, K=384, Nout=32) : 2 waves/block -> 1 N-group
    gemm_wmma_f32<<<dim3(mTiles, 1), 64, 0, stream>>>(OUT1, W2, H2, N, F1, OUT_DIM);

    attn_coef_kernel<<<blocks(N * 32), TB, 0, stream>>>(
        H2, att_src2, att_dst2, AS2, AD2, N, 1, OUT_DIM);

    fill_i32_kernel<<<blocks(N), TB, 0, stream>>>(M2, ENC_NEG_INF, N);
    fill_f32_kernel<<<blocks(N), TB, 0, stream>>>(DEN2, 0.f, N);
    fill_f32_kernel<<<blocks(N * OUT_DIM), TB, 0, stream>>>(OUT2, 0.f, N * OUT_DIM);

    edge_max_kernel<<<blocks(eTot), TB, 0, stream>>>(ei, E, eTot, AS2, AD2, 1, M2);
    edge_expsum_kernel<<<blocks(eTot), TB, 0, stream>>>(
        ei, E, eTot, AS2, AD2, 1, M2, EX2, DEN2);

    // 8 edges per 256-thread block (1 head x 32 ch)
    edge_aggregate_kernel<<<(eTot + 7) / 8, 256, 0, stream>>>(
        ei, E, eTot, H2, EX2, DEN2, OUT2, 1, OUT_DIM);

    bias_act_kernel<<<blocks(N * OUT_DIM), TB, 0, stream>>>(
        OUT2, b2, N * OUT_DIM, OUT_DIM, 0);
}